// GNNClassifier_22247930594079
// MI455X (gfx1250) — compile-verified
//
#include <hip/hip_runtime.h>
#include <math.h>

// ---------------------------------------------------------------------------
// GNN classifier for MI455X (gfx1250, wave32).
// Dense layers: V_WMMA_F32_16X16X4_F32, fully unrolled K-chains, B-tile held
// in registers across a grid-stride M loop. Edge scatter / pooling use
// hardware global_atomic_add_f32 (relaxed, agent scope); the whole feature
// working set (~51 MB) is L2-resident on the 192 MB L2.
// ---------------------------------------------------------------------------

typedef __attribute__((ext_vector_type(2))) float v2f;
typedef __attribute__((ext_vector_type(8))) float v8f;

#define N_NODES 100000
#define N_EDGES 1600000
#define N_TOKENS 16
#define TOK_DIM 64
#define NODE_DIM 64
#define HID 128
#define FC_DIM 128
#define NUM_GRAPHS 1024

__device__ __forceinline__ void atomic_add_f32(float* p, float v) {
    __hip_atomic_fetch_add(p, v, __ATOMIC_RELAXED, __HIP_MEMORY_SCOPE_AGENT);
}

// -------------------- 1. token embedding gather + mean ---------------------
// thread = (node, dim-quad). pooled[n, d] = mean_t emb[tok[n,t], d]
__global__ __launch_bounds__(256) void embed_pool_kernel(
    const int* __restrict__ tok, const float* __restrict__ emb,
    float* __restrict__ pooled) {
    int idx = blockIdx.x * blockDim.x + threadIdx.x;
    if (idx >= N_NODES * (TOK_DIM / 4)) return;
    int node = idx >> 4;          // TOK_DIM/4 == 16 quads
    int d4 = idx & 15;
    const int* ti = tok + node * N_TOKENS;
    float ax = 0.f, ay = 0.f, az = 0.f, aw = 0.f;
    for (int t = 0; t < N_TOKENS; ++t) {
        const float4 e = *(const float4*)(emb + (size_t)ti[t] * TOK_DIM + d4 * 4);
        ax += e.x; ay += e.y; az += e.z; aw += e.w;
    }
    const float s = 1.0f / 16.0f;
    float4 o; o.x = ax * s; o.y = ay * s; o.z = az * s; o.w = aw * s;
    *(float4*)(pooled + (size_t)node * TOK_DIM + d4 * 4) = o;
}

// -------------------- 2. WMMA f32 GEMM: C = A[M,K] @ W[K,NC] (+bias,+relu) --
// One wave per 16x16 output tile column; blockDim.x = 32*(NC/16).
// Grid-strides over M tiles with the K x 16 B-tile held in registers.
// A-operand (16x4 f32): lanes 0-15 M=lane; VGPR0={K+0|K+2}, VGPR1={K+1|K+3}
// B-operand (4x16 f32): lanes 0-15 N=lane; VGPR0={K+0|K+2}, VGPR1={K+1|K+3}
// C layout: VGPR r -> row (r + 8*half), col = lane&15.
template <int K, int NC, bool BIAS, bool RELU_OUT>
__global__ void wmma_gemm_kernel(const float* __restrict__ A,
                                 const float* __restrict__ W,
                                 const float* __restrict__ bias,
                                 float* __restrict__ Cout, int Mtiles) {
    const int lane = threadIdx.x & 31;
    const int wid  = threadIdx.x >> 5;          // n-tile index
    const int nbase = wid * 16;
    const int m = lane & 15;
    const int n = lane & 15;
    const int half = lane >> 4;

    // Preload B tile (K x 16), loop-invariant across all M tiles.
    v2f breg[K / 4];
    #pragma unroll
    for (int kk = 0; kk < K; kk += 4) {
        const int ka = kk + 2 * half;
        breg[kk / 4].x = W[(size_t)(ka + 0) * NC + nbase + n];
        breg[kk / 4].y = W[(size_t)(ka + 1) * NC + nbase + n];
    }
    float bv = 0.f;
    if (BIAS) bv = bias[nbase + n];

    for (int mt = blockIdx.x; mt < Mtiles; mt += gridDim.x) {
        const int mbase = mt * 16;
        const float* arow = A + (size_t)(mbase + m) * K + 2 * half;
        v8f acc = {};
        #pragma unroll
        for (int kk = 0; kk < K; kk += 4) {
            v2f a = *(const v2f*)(arow + kk);   // 8B aligned
            acc = __builtin_amdgcn_wmma_f32_16x16x4_f32(
                /*neg_a=*/false, a, /*neg_b=*/false, breg[kk / 4],
                /*c_mod=*/(short)0, acc, /*reuse_a=*/false, /*reuse_b=*/false);
        }
        #pragma unroll
        for (int r = 0; r < 8; ++r) {
            float v = acc[r] + bv;
            if (RELU_OUT) v = fmaxf(v, 0.f);
            Cout[(size_t)(mbase + r + 8 * half) * NC + nbase + n] = v;
        }
    }
}

// -------------------- 3. degree / rsqrt ------------------------------------
__global__ __launch_bounds__(256) void deg_init_kernel(float* __restrict__ deg) {
    int i = blockIdx.x * blockDim.x + threadIdx.x;
    if (i < N_NODES) deg[i] = 1.0f;                  // self loop
}
__global__ __launch_bounds__(256) void deg_count_kernel(
    const int* __restrict__ dst, float* __restrict__ deg) {
    int e = blockIdx.x * blockDim.x + threadIdx.x;
    if (e < N_EDGES) atomic_add_f32(&deg[dst[e]], 1.0f);
}
__global__ __launch_bounds__(256) void deg_rsqrt_kernel(float* __restrict__ deg) {
    int i = blockIdx.x * blockDim.x + threadIdx.x;
    if (i < N_NODES) deg[i] = rsqrtf(deg[i]);        // deg >= 1 always
}

// -------------------- 4. self-loop + bias init: x = b + dis^2 * h ----------
__global__ __launch_bounds__(256) void scatter_init_kernel(
    const float* __restrict__ h, const float* __restrict__ dis,
    const float* __restrict__ bias, float* __restrict__ x) {
    int idx = blockIdx.x * blockDim.x + threadIdx.x;
    if (idx >= N_NODES * (HID / 4)) return;
    int i = idx >> 5;                   // HID/4 == 32 quads per row
    int c4 = (idx & 31) * 4;
    float w = dis[i]; w = w * w;
    float4 hv = *(const float4*)(h + (size_t)i * HID + c4);
    float4 o;
    o.x = bias[c4 + 0] + w * hv.x;
    o.y = bias[c4 + 1] + w * hv.y;
    o.z = bias[c4 + 2] + w * hv.z;
    o.w = bias[c4 + 3] + w * hv.w;
    *(float4*)(x + (size_t)i * HID + c4) = o;
}

// -------------------- 5. edge scatter: x[dst] += dis[s]*dis[d]*h[src] ------
__global__ __launch_bounds__(256) void edge_scatter_kernel(
    const int* __restrict__ src, const int* __restrict__ dst,
    const float* __restrict__ dis, const float* __restrict__ h,
    float* __restrict__ x) {
    int idx = blockIdx.x * blockDim.x + threadIdx.x;
    if (idx >= N_EDGES * (HID / 4)) return;
    int e = idx >> 5;                   // 32 quads per edge row
    int c4 = (idx & 31) * 4;
    int s = src[e], d = dst[e];
    float w = dis[s] * dis[d];
    float4 hv = *(const float4*)(h + (size_t)s * HID + c4);
    float* xp = x + (size_t)d * HID + c4;
    atomic_add_f32(xp + 0, w * hv.x);
    atomic_add_f32(xp + 1, w * hv.y);
    atomic_add_f32(xp + 2, w * hv.z);
    atomic_add_f32(xp + 3, w * hv.w);
}

// -------------------- 6. relu ----------------------------------------------
__global__ __launch_bounds__(256) void relu_kernel(float* __restrict__ x, int n4) {
    int idx = blockIdx.x * blockDim.x + threadIdx.x;
    if (idx >= n4) return;
    float4 v = *(float4*)(x + (size_t)idx * 4);
    v.x = fmaxf(v.x, 0.f); v.y = fmaxf(v.y, 0.f);
    v.z = fmaxf(v.z, 0.f); v.w = fmaxf(v.w, 0.f);
    *(float4*)(x + (size_t)idx * 4) = v;
}

// -------------------- 7. graph mean pool -----------------------------------
__global__ __launch_bounds__(256) void pool_zero_kernel(
    float* __restrict__ gsum, float* __restrict__ gcnt) {
    int i = blockIdx.x * blockDim.x + threadIdx.x;
    if (i < NUM_GRAPHS * HID) gsum[i] = 0.f;
    if (i < NUM_GRAPHS) gcnt[i] = 0.f;
}
__global__ __launch_bounds__(256) void pool_cnt_kernel(
    const int* __restrict__ batch, float* __restrict__ gcnt) {
    int i = blockIdx.x * blockDim.x + threadIdx.x;
    if (i < N_NODES) atomic_add_f32(&gcnt[batch[i]], 1.0f);
}
__global__ __launch_bounds__(256) void pool_sum_kernel(
    const int* __restrict__ batch, const float* __restrict__ x,
    float* __restrict__ gsum) {
    int idx = blockIdx.x * blockDim.x + threadIdx.x;
    if (idx >= N_NODES * (HID / 4)) return;
    int i = idx >> 5;                   // HID/4 == 32 quads per node
    int c4 = (idx & 31) * 4;
    int g = batch[i];
    float4 v = *(const float4*)(x + (size_t)i * HID + c4);
    float* gp = gsum + (size_t)g * HID + c4;
    atomic_add_f32(gp + 0, v.x);
    atomic_add_f32(gp + 1, v.y);
    atomic_add_f32(gp + 2, v.z);
    atomic_add_f32(gp + 3, v.w);
}
__global__ __launch_bounds__(256) void pool_mean_kernel(
    const float* __restrict__ gsum, const float* __restrict__ gcnt,
    float* __restrict__ gmean) {
    int idx = blockIdx.x * blockDim.x + threadIdx.x;
    if (idx >= NUM_GRAPHS * (HID / 4)) return;
    int g = idx >> 5;
    int c4 = (idx & 31) * 4;
    float inv = 1.0f / fmaxf(gcnt[g], 1.0f);
    float4 v = *(const float4*)(gsum + (size_t)g * HID + c4);
    v.x *= inv; v.y *= inv; v.z *= inv; v.w *= inv;
    *(float4*)(gmean + (size_t)g * HID + c4) = v;
}

// -------------------- 8. head: logits = gfc @ Wout + bout; sigmoid ---------
__global__ __launch_bounds__(256) void head_kernel(
    const float* __restrict__ gfc, const float* __restrict__ Wout,
    const float* __restrict__ bout, float* __restrict__ out) {
    int g = blockIdx.x * blockDim.x + threadIdx.x;
    if (g >= NUM_GRAPHS) return;
    const float* row = gfc + (size_t)g * FC_DIM;
    float acc = 0.f;
    for (int k = 0; k < FC_DIM; ++k) acc += row[k] * Wout[k];
    float logit = acc + bout[0];
    out[g] = 1.0f / (1.0f + expf(-logit));           // sigmoid
    out[NUM_GRAPHS + g] = logit;                     // raw logits
}

// ---------------------------------------------------------------------------
extern "C" void kernel_launch(void* const* d_in, const int* in_sizes, int n_in,
                              void* d_out, int out_size, void* d_ws, size_t ws_size,
                              hipStream_t stream) {
    const int*   tok   = (const int*)d_in[0];    // [N, 16]
    const int*   edges = (const int*)d_in[1];    // [2, E]
    const int*   batch = (const int*)d_in[2];    // [N]
    const float* emb   = (const float*)d_in[3];  // [32000, 64]
    const float* Wp    = (const float*)d_in[4];
    const float* bp    = (const float*)d_in[5];
    const float* W1    = (const float*)d_in[6];
    const float* b1    = (const float*)d_in[7];
    const float* W2    = (const float*)d_in[8];
    const float* b2    = (const float*)d_in[9];
    const float* Wfc   = (const float*)d_in[10];
    const float* bfc   = (const float*)d_in[11];
    const float* Wout  = (const float*)d_in[12];
    const float* bout  = (const float*)d_in[13];
    float* out = (float*)d_out;

    const int* e_src = edges;
    const int* e_dst = edges + N_EDGES;

    // workspace layout (floats); total ~130 MB
    float* ws = (float*)d_ws;
    size_t off = 0;
    float* pooled = ws + off; off += (size_t)N_NODES * TOK_DIM;   // [N,64]
    float* B1     = ws + off; off += (size_t)N_NODES * HID;       // node / x1 / x2
    float* B2     = ws + off; off += (size_t)N_NODES * HID;       // h1 / h2
    float* dis    = ws + off; off += N_NODES;                     // rsqrt(deg)
    float* gsum   = ws + off; off += (size_t)NUM_GRAPHS * HID;
    float* gcnt   = ws + off; off += NUM_GRAPHS;
    float* gmean  = ws + off; off += (size_t)NUM_GRAPHS * HID;
    float* gfc    = ws + off; off += (size_t)NUM_GRAPHS * FC_DIM;
    (void)ws_size; (void)n_in; (void)in_sizes; (void)out_size;

    const int T = 256;
    const int MT_N = N_NODES / 16;      // 6250 tiles (exact)
    const int MT_G = NUM_GRAPHS / 16;   // 64 tiles (exact)
    const int GEMM_BLK = 1024;          // grid-stride blocks for node-level GEMMs

    // 1. embedding gather + mean over tokens
    embed_pool_kernel<<<(N_NODES * 16 + T - 1) / T, T, 0, stream>>>(tok, emb, pooled);

    // 2. node = pooled @ Wp + bp      [N,64] (stored stride 64 in B1)
    wmma_gemm_kernel<TOK_DIM, NODE_DIM, true, false>
        <<<GEMM_BLK, 32 * (NODE_DIM / 16), 0, stream>>>(pooled, Wp, bp, B1, MT_N);

    // 3. symmetric GCN norm: dis = rsqrt(1 + indeg)
    deg_init_kernel<<<(N_NODES + T - 1) / T, T, 0, stream>>>(dis);
    deg_count_kernel<<<(N_EDGES + T - 1) / T, T, 0, stream>>>(e_dst, dis);
    deg_rsqrt_kernel<<<(N_NODES + T - 1) / T, T, 0, stream>>>(dis);

    // ---- GCN layer 1: x1 = relu(scatter(node @ W1) + b1) ----
    wmma_gemm_kernel<NODE_DIM, HID, false, false>
        <<<GEMM_BLK, 32 * (HID / 16), 0, stream>>>(B1, W1, nullptr, B2, MT_N);  // h1 -> B2
    scatter_init_kernel<<<(N_NODES * (HID / 4) + T - 1) / T, T, 0, stream>>>(
        B2, dis, b1, B1);                                     // x1 <- b1 + dis^2*h1
    edge_scatter_kernel<<<(N_EDGES * (HID / 4) + T - 1) / T, T, 0, stream>>>(
        e_src, e_dst, dis, B2, B1);
    relu_kernel<<<(N_NODES * (HID / 4) + T - 1) / T, T, 0, stream>>>(
        B1, N_NODES * (HID / 4));

    // ---- GCN layer 2: x2 = relu(scatter(x1 @ W2) + b2) ----
    wmma_gemm_kernel<HID, HID, false, false>
        <<<GEMM_BLK, 32 * (HID / 16), 0, stream>>>(B1, W2, nullptr, B2, MT_N);  // h2 -> B2
    scatter_init_kernel<<<(N_NODES * (HID / 4) + T - 1) / T, T, 0, stream>>>(
        B2, dis, b2, B1);                                     // x2 <- b2 + dis^2*h2
    edge_scatter_kernel<<<(N_EDGES * (HID / 4) + T - 1) / T, T, 0, stream>>>(
        e_src, e_dst, dis, B2, B1);
    relu_kernel<<<(N_NODES * (HID / 4) + T - 1) / T, T, 0, stream>>>(
        B1, N_NODES * (HID / 4));

    // ---- global mean pool ----
    pool_zero_kernel<<<(NUM_GRAPHS * HID + T - 1) / T, T, 0, stream>>>(gsum, gcnt);
    pool_cnt_kernel<<<(N_NODES + T - 1) / T, T, 0, stream>>>(batch, gcnt);
    pool_sum_kernel<<<(N_NODES * (HID / 4) + T - 1) / T, T, 0, stream>>>(batch, B1, gsum);
    pool_mean_kernel<<<(NUM_GRAPHS * (HID / 4) + T - 1) / T, T, 0, stream>>>(gsum, gcnt, gmean);

    // ---- FC: gfc = relu(gmean @ Wfc + bfc) ----
    wmma_gemm_kernel<HID, FC_DIM, true, true>
        <<<MT_G, 32 * (FC_DIM / 16), 0, stream>>>(gmean, Wfc, bfc, gfc, MT_G);

    // ---- head: logits + sigmoid ----
    head_kernel<<<(NUM_GRAPHS + T - 1) / T, T, 0, stream>>>(gfc, Wout, bout, out);
}